// GCNN_4277787427600
// MI455X (gfx1250) — compile-verified
//
#include <hip/hip_runtime.h>
#include <hip/hip_bf16.h>

// ---------------------------------------------------------------------------
// GCN forward on MI455X (gfx1250, wave32).
//   h = x @ W1 done with V_WMMA_F32_16X16X4_F32 (full f32 precision),
//   B fragments staged zero-padded + transposed in LDS (ds_load_b64, no
//   divergent exec masking in the K loop).
//   Edge scatter / pooling are L2-resident float atomics (h/acc = 16MB each,
//   well inside the 192MB L2; edge stream ~32MB from HBM @ 23.3 TB/s).
// ---------------------------------------------------------------------------

#define NNODES 200000
#define NEDGES 4000000
#define NGRAPH 512
#define DIM    20
#define DOUT   5

typedef float v2f __attribute__((ext_vector_type(2)));
typedef float v8f __attribute__((ext_vector_type(8)));

// ---------------- init: deg=1 (self loop), acc=0, sums=0, cnt=0 -------------
__global__ __launch_bounds__(256) void k_init(float* __restrict__ deg,
                                              float* __restrict__ acc,
                                              float* __restrict__ sums,
                                              float* __restrict__ cnt) {
    int i = blockIdx.x * blockDim.x + threadIdx.x;
    if (i < NNODES * DIM) acc[i] = 0.0f;
    if (i < NNODES)       deg[i] = 1.0f;          // self loop contributes 1
    if (i < NGRAPH * DIM) sums[i] = 0.0f;
    if (i < NGRAPH)       cnt[i]  = 0.0f;
}

// ---------------- degree: deg[dst] += 1 per edge ----------------------------
__global__ __launch_bounds__(256) void k_degree(const int* __restrict__ edst,
                                                float* __restrict__ deg) {
    int e = blockIdx.x * blockDim.x + threadIdx.x;
    if (e < NEDGES) atomicAdd(&deg[edst[e]], 1.0f);
}

// ---------------- dinv = rsqrt(deg) in place (deg >= 1 always) --------------
__global__ __launch_bounds__(256) void k_rsqrt(float* __restrict__ deg) {
    int i = blockIdx.x * blockDim.x + threadIdx.x;
    if (i < NNODES) deg[i] = rsqrtf(deg[i]);
}

// ---------------- h = x @ W1 via V_WMMA_F32_16X16X4_F32 ---------------------
// One wave per 16-row block. Two 16x16 output tiles (cols 0-15, 16-31 with
// zero padding). K = 20 = 5 steps of 4.
// Weights staged in LDS transposed + padded: Wt[n][k], n<32, k<24 (zeros
// outside 20x20) so each B fragment is a contiguous aligned b64 LDS load.
// A 16x4 f32 layout:  lanes 0-15: v0=K0,v1=K1 ; lanes 16-31: v0=K2,v1=K3 ; M=lane&15
// C   16x16 f32:      lane&15 = N ; VGPR r -> M = r + 8*(lane>=16)
#define WT_NP 32   // padded N
#define WT_KP 24   // padded K (even, keeps 8B alignment for (n*WT_KP+kb)*4)
__global__ __launch_bounds__(256) void k_gemm_wmma(const float* __restrict__ x,
                                                   const float* __restrict__ W,
                                                   float* __restrict__ h,
                                                   int nRowBlocks) {
    __shared__ float Wt[WT_NP * WT_KP];          // 3 KB
    for (int idx = threadIdx.x; idx < WT_NP * WT_KP; idx += 256) {
        const int n = idx / WT_KP;
        const int k = idx % WT_KP;
        Wt[idx] = (n < DIM && k < DIM) ? W[k * DIM + n] : 0.0f;
    }
    __syncthreads();

    const int lane = threadIdx.x & 31;
    const int wave = threadIdx.x >> 5;
    const int rb   = blockIdx.x * 8 + wave;      // wave-uniform
    if (rb >= nRowBlocks) return;                // whole wave exits -> EXEC all-1s

    const int row0 = rb * 16;
    const int half = lane >> 4;                  // K-pair select
    const int m    = lane & 15;                  // M for A, N for B/C/D
    const int n1   = 16 + m;                     // second-tile column

    const float* __restrict__ arow = x  + (size_t)(row0 + m) * DIM;
    const float* __restrict__ wt0  = Wt + m  * WT_KP;
    const float* __restrict__ wt1  = Wt + n1 * WT_KP;

    v8f c0 = {};
    v8f c1 = {};
#pragma unroll
    for (int k0 = 0; k0 < DIM; k0 += 4) {
        const int kb = k0 + 2 * half;            // this lane's K base (kb, kb+1)
        const v2f a  = *(const v2f*)(arow + kb); // 8B-aligned global_load_b64
        const v2f b0 = *(const v2f*)(wt0 + kb);  // 8B-aligned ds_load_b64
        const v2f b1 = *(const v2f*)(wt1 + kb);

        c0 = __builtin_amdgcn_wmma_f32_16x16x4_f32(false, a, false, b0,
                                                   (short)0, c0, false, false);
        c1 = __builtin_amdgcn_wmma_f32_16x16x4_f32(false, a, false, b1,
                                                   (short)0, c1, false, false);
    }

#pragma unroll
    for (int r = 0; r < 8; ++r) {
        const int row = row0 + r + 8 * half;
        h[(size_t)row * DIM + m] = c0[r];
        if (n1 < DIM) h[(size_t)row * DIM + n1] = c1[r];
    }
}

// ---------------- edge scatter: acc[dst] += h[src] * dinv[src]*dinv[dst] ----
// 5 threads per edge, one aligned float4 (16B) chunk of the 80B row each.
__global__ __launch_bounds__(256) void k_edge_scatter(const int* __restrict__ esrc,
                                                      const int* __restrict__ edst,
                                                      const float* __restrict__ h,
                                                      const float* __restrict__ dinv,
                                                      float* __restrict__ acc) {
    long long t = (long long)blockIdx.x * blockDim.x + threadIdx.x;
    if (t >= (long long)NEDGES * 5) return;
    const int e = (int)(t / 5);
    const int p = (int)(t % 5);

    const int s = esrc[e];
    const int d = edst[e];
    const float w = dinv[s] * dinv[d];

    const float4 v = *(const float4*)(h + (size_t)s * DIM + p * 4);
    float* ar = acc + (size_t)d * DIM + p * 4;
    atomicAdd(ar + 0, v.x * w);
    atomicAdd(ar + 1, v.y * w);
    atomicAdd(ar + 2, v.z * w);
    atomicAdd(ar + 3, v.w * w);
}

// ---------------- bias + self-loop + relu + mean-pool accumulate ------------
__global__ __launch_bounds__(256) void k_relu_pool(const float* __restrict__ acc,
                                                   const float* __restrict__ h,
                                                   const float* __restrict__ dinv,
                                                   const float* __restrict__ b1,
                                                   const int* __restrict__ batch,
                                                   float* __restrict__ sums,
                                                   float* __restrict__ cnt) {
    int i = blockIdx.x * blockDim.x + threadIdx.x;
    if (i >= NNODES) return;
    const float di = dinv[i];
    const float w  = di * di;                    // self-loop norm
    const int   g  = batch[i];
    const float* hr = h   + (size_t)i * DIM;
    const float* ar = acc + (size_t)i * DIM;
    float* sg = sums + (size_t)g * DIM;
#pragma unroll
    for (int d = 0; d < DIM; ++d) {
        float v = ar[d] + hr[d] * w + b1[d];
        v = v > 0.0f ? v : 0.0f;
        atomicAdd(sg + d, v);
    }
    atomicAdd(cnt + g, 1.0f);
}

// ---------------- head: mean, 20->5 linear, softmax -------------------------
__global__ __launch_bounds__(256) void k_head(const float* __restrict__ sums,
                                              const float* __restrict__ cnt,
                                              const float* __restrict__ Wout,
                                              const float* __restrict__ bout,
                                              float* __restrict__ out) {
    int g = blockIdx.x * blockDim.x + threadIdx.x;
    if (g >= NGRAPH) return;
    float c = cnt[g];
    c = c > 1.0f ? c : 1.0f;
    const float inv = 1.0f / c;

    float logits[DOUT];
#pragma unroll
    for (int o = 0; o < DOUT; ++o) logits[o] = bout[o];
#pragma unroll
    for (int d = 0; d < DIM; ++d) {
        const float p = sums[(size_t)g * DIM + d] * inv;
#pragma unroll
        for (int o = 0; o < DOUT; ++o) logits[o] += p * Wout[d * DOUT + o];
    }
    float mx = logits[0];
#pragma unroll
    for (int o = 1; o < DOUT; ++o) mx = fmaxf(mx, logits[o]);
    float ex[DOUT]; float se = 0.0f;
#pragma unroll
    for (int o = 0; o < DOUT; ++o) { ex[o] = expf(logits[o] - mx); se += ex[o]; }
    const float ise = 1.0f / se;
#pragma unroll
    for (int o = 0; o < DOUT; ++o) out[(size_t)g * DOUT + o] = ex[o] * ise;
}

// ---------------------------------------------------------------------------
extern "C" void kernel_launch(void* const* d_in, const int* in_sizes, int n_in,
                              void* d_out, int out_size, void* d_ws, size_t ws_size,
                              hipStream_t stream) {
    (void)in_sizes; (void)n_in; (void)out_size; (void)ws_size;

    const float* x     = (const float*)d_in[0];
    const int*   eidx  = (const int*)  d_in[1];   // (2, E): [src | dst]
    const int*   batch = (const int*)  d_in[2];
    const float* W1    = (const float*)d_in[3];
    const float* b1    = (const float*)d_in[4];
    const float* Wout  = (const float*)d_in[5];
    const float* bout  = (const float*)d_in[6];
    float*       out   = (float*)d_out;

    const int* esrc = eidx;
    const int* edst = eidx + NEDGES;

    // Workspace layout (floats): deg/dinv | h | acc | sums | cnt  (~33 MB)
    float* f    = (float*)d_ws;
    float* deg  = f;                                   // NNODES  (becomes dinv)
    float* h    = f + NNODES;                          // NNODES*DIM (16B aligned)
    float* acc  = f + NNODES + NNODES * DIM;           // NNODES*DIM
    float* sums = f + NNODES + 2 * NNODES * DIM;       // NGRAPH*DIM
    float* cnt  = sums + NGRAPH * DIM;                 // NGRAPH

    const int TPB = 256;

    k_init   <<<(NNODES * DIM + TPB - 1) / TPB, TPB, 0, stream>>>(deg, acc, sums, cnt);
    k_degree <<<(NEDGES + TPB - 1) / TPB, TPB, 0, stream>>>(edst, deg);
    k_rsqrt  <<<(NNODES + TPB - 1) / TPB, TPB, 0, stream>>>(deg);

    const int nRowBlocks = NNODES / 16;                // 12500 exactly
    k_gemm_wmma<<<(nRowBlocks + 7) / 8, TPB, 0, stream>>>(x, W1, h, nRowBlocks);

    const long long T = (long long)NEDGES * 5;         // 20M threads
    k_edge_scatter<<<(int)((T + TPB - 1) / TPB), TPB, 0, stream>>>(esrc, edst, h, deg, acc);

    k_relu_pool<<<(NNODES + TPB - 1) / TPB, TPB, 0, stream>>>(acc, h, deg, b1, batch, sums, cnt);
    k_head     <<<(NGRAPH + TPB - 1) / TPB, TPB, 0, stream>>>(sums, cnt, Wout, bout, out);
}